// ChannelwiseClassifier_41678362640503
// MI455X (gfx1250) — compile-verified
//
#include <hip/hip_runtime.h>
#include <hip/hip_bf16.h>
#include <cstdint>
#include <cstddef>

// Problem constants (reference: N=4096, N_WAY=32, FEAT_DIM=1024)
#define NWAY 32
#define FDIM 1024
#define TPB  256          // 8 wave32s
#define ROWS_PER_WG 8     // amortize the 128KB W staging across rows

typedef __attribute__((ext_vector_type(2)))  float v2f;
typedef __attribute__((ext_vector_type(8)))  float v8f;
typedef __attribute__((ext_vector_type(4)))  int   v4i;

#define GLOBAL_AS __attribute__((address_space(1)))
#define LDS_AS    __attribute__((address_space(3)))

#define LOG2E_F 1.4426950408889634f
#define LN2_F   0.6931471805599453f

// CDNA5 async global->LDS path (falls back to register copy if the builtin is
// absent on this toolchain / on the host compilation pass).
#if __has_builtin(__builtin_amdgcn_global_load_async_to_lds_b128) && \
    __has_builtin(__builtin_amdgcn_s_wait_asynccnt)
#define USE_ASYNC_LDS 1
#else
#define USE_ASYNC_LDS 0
#endif

__device__ __forceinline__ float fast_exp2(float x) {
#if __has_builtin(__builtin_amdgcn_exp2f)
  return __builtin_amdgcn_exp2f(x);   // v_exp_f32
#else
  return exp2f(x);
#endif
}

__global__ __launch_bounds__(TPB) void cwcls_kernel(
    const float* __restrict__ X, const float* __restrict__ W,
    float* __restrict__ prob, float* __restrict__ soft, int nrows)
{
  __shared__ float lds_w[NWAY * FDIM];   // 128 KB: W * log2e, row-major [c][d]
  __shared__ float lds_p8[8][33];        // 8 padded partial rows (stride 33 ->
                                         // rows hit 8 distinct LDS banks)

  const int tid  = threadIdx.x;
  const int lane = tid & 31;

  // ---------------- Stage W (32x1024 f32 = 128 KB) into LDS ----------------
#if USE_ASYNC_LDS
  {
    // 256 lanes x 16B x 32 iterations = 131072 B, ASYNCcnt-tracked, no VGPR
    // staging. Builtin signature: (v4i AS1*, v4i AS3*, imm offset, imm cpol).
    #pragma unroll
    for (int it = 0; it < (NWAY * FDIM * 4) / (TPB * 16); ++it) {
      const int boff = (it * TPB + tid) * 16;
      __builtin_amdgcn_global_load_async_to_lds_b128(
          (GLOBAL_AS v4i*)((const char*)W + boff),
          (LDS_AS v4i*)((char*)lds_w + boff),
          /*offset=*/0, /*cpol=*/0);
    }
    __builtin_amdgcn_s_wait_asynccnt(0);
  }
  __syncthreads();
  // In-place pre-scale by log2e so the hot loop works in the exp2 domain:
  // per class it then costs only fma + mul + v_exp_f32.
  #pragma unroll
  for (int e4 = tid * 4; e4 < NWAY * FDIM; e4 += TPB * 4) {
    float4 v = *(float4*)(lds_w + e4);
    v.x *= LOG2E_F; v.y *= LOG2E_F; v.z *= LOG2E_F; v.w *= LOG2E_F;
    *(float4*)(lds_w + e4) = v;
  }
#else
  for (int e4 = tid * 4; e4 < NWAY * FDIM; e4 += TPB * 4) {
    float4 v = *(const float4*)(W + e4);
    v.x *= LOG2E_F; v.y *= LOG2E_F; v.z *= LOG2E_F; v.w *= LOG2E_F;
    *(float4*)(lds_w + e4) = v;
  }
#endif
  __syncthreads();

  for (int r = 0; r < ROWS_PER_WG; ++r) {
    const int row = blockIdx.x * ROWS_PER_WG + r;
    if (row >= nrows) break;

    for (int i = tid; i < 8 * 33; i += TPB) ((float*)lds_p8)[i] = 0.0f;
    __syncthreads();

    float acc[NWAY];
    #pragma unroll
    for (int c = 0; c < NWAY; ++c) acc[c] = 0.0f;

    const float* Xrow = X + (size_t)row * FDIM;
    float*       Srow = soft + (size_t)row * (size_t)(NWAY * FDIM);

    // Each thread owns d = tid, tid+256, ... (4 columns). Lanes have
    // consecutive d -> 128B-coalesced global traffic per wave and
    // bank-conflict-free lds_w[c*FDIM+d] reads.
    for (int d = tid; d < FDIM; d += TPB) {
      const float x2 = 2.0f * Xrow[d];

      // log2-domain logit: t = log2e*(2xw - w^2) = s*(x2 - s*ln2), s = w*log2e.
      // (-(w-x)^2 differs only by the class-constant x^2, which cancels in
      //  softmax; no max-subtraction needed: |t| <~ 160, safely inside f32 exp2.)
      float e[NWAY];
      float s = 0.0f;
      #pragma unroll
      for (int c = 0; c < NWAY; ++c) {
        const float sl = lds_w[c * FDIM + d];
        const float t  = sl * __builtin_fmaf(sl, -LN2_F, x2);
        const float p  = fast_exp2(t);
        e[c] = p;
        s += p;
      }
      const float inv = 1.0f / s;

      // 537MB write-once stream: non-temporal so it does not thrash L2.
      #pragma unroll
      for (int c = 0; c < NWAY; ++c) {
        const float p = e[c] * inv;
        __builtin_nontemporal_store(p, Srow + c * FDIM + d);
        acc[c] += p;
      }
    }

    // ---- reduce acc[32] over 256 threads -> 8 partial rows in LDS ----
    // Static register index (no cndmask chains); lane -> padded row (l&7),
    // rows land in 8 distinct banks, only a 4-way same-address RMW per op.
    {
      const int prow = lane & 7;
      #pragma unroll
      for (int c = 0; c < NWAY; ++c)
        atomicAdd(&lds_p8[prow][c], acc[c]);
    }
    __syncthreads();

    // ---- final 8x32 column sum on wave 0 via two chained WMMAs per half ----
    // D = A(16x4,ones) * B(4x16) + C  ->  every row of D = colsum(B) + C.
    // A = all-ones is layout-independent; C/D layout: VGPR0 lanes0-15 = row M=0.
#if __has_builtin(__builtin_amdgcn_wmma_f32_16x16x4_f32)
    if (tid < 32) {                       // wave 0, EXEC all-ones inside
      v2f a; a[0] = 1.0f; a[1] = 1.0f;    // ones(16x4)
      #pragma unroll
      for (int h = 0; h < 2; ++h) {
        const int n  = (lane & 15) + 16 * h;
        const int kb = (lane < 16) ? 0 : 2;   // B f32 4x16: v0={K0|K2}, v1={K1|K3}
        v2f bhi, blo;
        bhi[0] = lds_p8[kb + 4][n]; bhi[1] = lds_p8[kb + 5][n];  // rows 4..7
        blo[0] = lds_p8[kb + 0][n]; blo[1] = lds_p8[kb + 1][n];  // rows 0..3
        v8f cd = {};
        cd = __builtin_amdgcn_wmma_f32_16x16x4_f32(
            false, a, false, bhi, (short)0, cd, false, false);
        cd = __builtin_amdgcn_wmma_f32_16x16x4_f32(
            false, a, false, blo, (short)0, cd, false, false);
        if (lane < 16)
          prob[(size_t)row * NWAY + n] = cd[0] * (1.0f / FDIM);
      }
    }
#else
    if (tid < NWAY) {
      float t = 0.0f;
      #pragma unroll
      for (int k = 0; k < 8; ++k) t += lds_p8[k][tid];
      prob[(size_t)row * NWAY + tid] = t * (1.0f / FDIM);
    }
#endif
    __syncthreads();   // protect lds_p8 before next row's zeroing
  }
}

extern "C" void kernel_launch(void* const* d_in, const int* in_sizes, int n_in,
                              void* d_out, int out_size, void* d_ws, size_t ws_size,
                              hipStream_t stream) {
  (void)n_in; (void)out_size; (void)d_ws; (void)ws_size;
  const float* X = (const float*)d_in[0];   // [N, FDIM] f32
  const float* W = (const float*)d_in[1];   // [NWAY, FDIM] f32
  const int n = in_sizes[0] / FDIM;         // 4096

  float* prob = (float*)d_out;                       // [N, NWAY]
  float* soft = prob + (size_t)n * NWAY;             // [N, NWAY, FDIM]

  const int grid = (n + ROWS_PER_WG - 1) / ROWS_PER_WG;  // 512
  hipLaunchKernelGGL(cwcls_kernel, dim3(grid), dim3(TPB), 0, stream,
                     X, W, prob, soft, n);
}